// CRF_48241072668831
// MI455X (gfx1250) — compile-verified
//
#include <hip/hip_runtime.h>
#include <math.h>

// DenseCRF mean-field on MI455X (gfx1250, wave32).
// fp32 end-to-end; working set is L2-resident (192MB); splat uses L2 f32 atomics.
// Compatibility transform (mu = COMPAT * I) implemented as a real
// V_WMMA_F32_16X16X4_F32 matmul fused with the softmax update.

#define C_CH 21

typedef __attribute__((ext_vector_type(2))) float v2f;
typedef __attribute__((ext_vector_type(8))) float v8f;

// ---------------------------------------------------------------- utilities
__global__ void zero_kernel(float* __restrict__ p, long n) {
  long i = (long)blockIdx.x * blockDim.x + threadIdx.x;
  if (i < n) p[i] = 0.0f;
}

// ------------------------------------------------------------------- splat
template<int J>
__global__ void splat_ones_kernel(const int* __restrict__ os,
                                  const float* __restrict__ ws,
                                  float* __restrict__ vals, int N) {
  int n = blockIdx.x * blockDim.x + threadIdx.x;
  if (n >= N) return;
#pragma unroll
  for (int j = 0; j < J; ++j) {
    int o = os[n * J + j] + 1;
    atomicAdd(&vals[o], ws[n * J + j]);
  }
}

template<int J>
__global__ void splat_kernel(const float* __restrict__ Q,
                             const int* __restrict__ os,
                             const float* __restrict__ ws,
                             float* __restrict__ vals, int N) {
  int t = blockIdx.x * blockDim.x + threadIdx.x;
  if (t >= N * C_CH) return;
  int n = t / C_CH, c = t % C_CH;
  float q = Q[t];
#pragma unroll
  for (int j = 0; j < J; ++j) {
    int o = os[n * J + j] + 1;
    float w = ws[n * J + j];
    atomicAdd(&vals[(long)o * C_CH + c], w * q);
  }
}

// -------------------------------------------------------------------- blur
// out[m+1] = in[m+1] + 0.5*(in[bn[m][0]] + in[bn[m][1]]);  row 0 stays zero pad.
template<int C>
__global__ void blur_kernel(const float* __restrict__ in,
                            float* __restrict__ out,
                            const int* __restrict__ bnj, int M) {
  long t = (long)blockIdx.x * blockDim.x + threadIdx.x;
  long total = (long)(M + 1) * C;
  if (t >= total) return;
  int row = (int)(t / C), c = (int)(t % C);
  if (row == 0) { out[t] = 0.0f; return; }
  int m = row - 1;
  int n1 = bnj[m * 2 + 0];
  int n2 = bnj[m * 2 + 1];
  out[t] = in[t] + 0.5f * (in[(long)n1 * C + c] + in[(long)n2 * C + c]);
}

// --------------------------------------------------------- slice / normalize
template<int J>
__global__ void inv_kernel(const int* __restrict__ os,
                           const float* __restrict__ ws,
                           const float* __restrict__ valsN,
                           float* __restrict__ inv, float alpha, int N) {
  int n = blockIdx.x * blockDim.x + threadIdx.x;
  if (n >= N) return;
  float s = 0.0f;
#pragma unroll
  for (int j = 0; j < J; ++j) s += ws[n * J + j] * valsN[os[n * J + j] + 1];
  inv[n] = 1.0f / (alpha * s + 1e-20f);
}

template<int J, bool ADD>
__global__ void slice_kernel(const int* __restrict__ os,
                             const float* __restrict__ ws,
                             const float* __restrict__ vals,
                             const float* __restrict__ inv,
                             float coef, float* __restrict__ msg, int N) {
  int t = blockIdx.x * blockDim.x + threadIdx.x;
  if (t >= N * C_CH) return;
  int n = t / C_CH, c = t % C_CH;
  float s = 0.0f;
#pragma unroll
  for (int j = 0; j < J; ++j)
    s += ws[n * J + j] * vals[(long)(os[n * J + j] + 1) * C_CH + c];
  float v = coef * inv[n] * s;
  if (ADD) msg[t] += v; else msg[t] = v;
}

// ----------------------------------------------------------------- softmax
__global__ void softmax_init_kernel(const float* __restrict__ U,
                                    float* __restrict__ Q, int N) {
  int n = blockIdx.x * blockDim.x + threadIdx.x;
  if (n >= N) return;
  const float* u = U + (long)n * C_CH;
  float v[C_CH];
  float mx = -3.4e38f;
#pragma unroll
  for (int c = 0; c < C_CH; ++c) { v[c] = -u[c]; mx = fmaxf(mx, v[c]); }
  float s = 0.0f;
#pragma unroll
  for (int c = 0; c < C_CH; ++c) { v[c] = __expf(v[c] - mx); s += v[c]; }
  float r = 1.0f / s;
  float* q = Q + (long)n * C_CH;
#pragma unroll
  for (int c = 0; c < C_CH; ++c) q[c] = v[c] * r;
}

// --------------------------------------------- WMMA compat + softmax update
// pairwise = Msg(16x21) x mu^T (mu = COMPAT * I), via V_WMMA_F32_16X16X4_F32.
// logits = -U - pairwise; Q = softmax(logits). One wave per 16-pixel tile.
__global__ void __launch_bounds__(256)
compat_softmax_wmma_kernel(const float* __restrict__ U,
                           const float* __restrict__ msg,
                           float* __restrict__ Q, int N) {
  __shared__ float tile[8][16][32];
  const int wslot = threadIdx.x >> 5;
  const int lane  = threadIdx.x & 31;
  const int p0 = ((int)blockIdx.x * 8 + wslot) * 16;
  if (p0 >= N) return;                       // wave-uniform; no barriers used

  const int half = lane >> 4;                // 0: K 0..1 | 1: K 2..3 (per chunk)
  const int l15  = lane & 15;
  const int kh   = half * 2;
  const float* mrow = msg + (long)(p0 + l15) * C_CH;   // A row = pixel l15
  const float MU = -1.0f;                    // COMPAT (Potts: mu = COMPAT * I)

  v8f acc0 = {};                             // D tile: channels 0..15
  v8f acc1 = {};                             // D tile: channels 16..31 (21 used)
#pragma unroll
  for (int k = 0; k < 24; k += 4) {          // K = 21 padded to 24
    int k0 = k + kh, k1 = k0 + 1;
    v2f a, b0, b1;
    a.x  = (k0 < C_CH) ? mrow[k0] : 0.0f;    // A 16x4: lane-half selects K pair
    a.y  = (k1 < C_CH) ? mrow[k1] : 0.0f;
    b0.x = (l15 == k0) ? MU : 0.0f;          // B 4x16: mu^T chunk, N-tile 0
    b0.y = (l15 == k1) ? MU : 0.0f;
    b1.x = (l15 + 16 == k0) ? MU : 0.0f;     // N-tile 1 (channels 16..)
    b1.y = (l15 + 16 == k1) ? MU : 0.0f;
    acc0 = __builtin_amdgcn_wmma_f32_16x16x4_f32(false, a, false, b0,
                                                 (short)0, acc0, false, false);
    acc1 = __builtin_amdgcn_wmma_f32_16x16x4_f32(false, a, false, b1,
                                                 (short)0, acc1, false, false);
  }

  // C/D layout: VGPR r holds row M = r + 8*half, col = l15 (+16 for tile 1).
#pragma unroll
  for (int r = 0; r < 8; ++r) {
    tile[wslot][r + 8 * half][l15]      = acc0[r];
    tile[wslot][r + 8 * half][16 + l15] = acc1[r];
  }
  // Same-wave LDS RAW: DS pipe is in-order per wave; no cross-wave sharing.

  if (lane < 16) {
    int p = p0 + lane;
    const float* u = U + (long)p * C_CH;
    float v[C_CH];
    float mx = -3.4e38f;
#pragma unroll
    for (int c = 0; c < C_CH; ++c) {
      v[c] = -u[c] - tile[wslot][lane][c];   // -U - COMPAT*msg
      mx = fmaxf(mx, v[c]);
    }
    float s = 0.0f;
#pragma unroll
    for (int c = 0; c < C_CH; ++c) { v[c] = __expf(v[c] - mx); s += v[c]; }
    float r = 1.0f / s;
    float* q = Q + (long)p * C_CH;
#pragma unroll
    for (int c = 0; c < C_CH; ++c) q[c] = v[c] * r;
  }
}

// ------------------------------------------------------------------- driver
extern "C" void kernel_launch(void* const* d_in, const int* in_sizes, int n_in,
                              void* d_out, int out_size, void* d_ws, size_t ws_size,
                              hipStream_t stream) {
  const float* unary = (const float*)d_in[0];
  const float* ws_b  = (const float*)d_in[1];
  const float* ws_s  = (const float*)d_in[2];
  const int*   os_b  = (const int*)d_in[3];
  const int*   os_s  = (const int*)d_in[4];
  const int*   bn_b  = (const int*)d_in[5];
  const int*   bn_s  = (const int*)d_in[6];

  const int N  = in_sizes[0] / C_CH;        // 262144
  const int Mb = in_sizes[5] / 12;          // bn_b: (6, Mb, 2)
  const int Ms = in_sizes[6] / 6;           // bn_s: (3, Ms, 2)
  const long Mmax = (Mb > Ms) ? Mb : Ms;

  float* Q    = (float*)d_ws;
  float* MSG  = Q    + (long)N * C_CH;
  float* INVB = MSG  + (long)N * C_CH;
  float* INVS = INVB + N;
  float* VA   = INVS + N;
  float* VB   = VA   + (Mmax + 1) * C_CH;

  const float a_b = 1.0f / (1.0f + 0.03125f);  // 1/(1+2^-5)
  const float a_s = 1.0f / (1.0f + 0.25f);     // 1/(1+2^-2)
  const int TB = 256;
  auto blk = [](long n) { return (int)((n + 255) / 256); };

  // ---- normalization constants: filter(ones) for both lattices (C=1) ----
  zero_kernel<<<blk(Mb + 1), TB, 0, stream>>>(VA, (long)(Mb + 1));
  splat_ones_kernel<6><<<blk(N), TB, 0, stream>>>(os_b, ws_b, VA, N);
  {
    float* in = VA; float* out = VB;
    for (int j = 0; j < 6; ++j) {
      blur_kernel<1><<<blk((long)Mb + 1), TB, 0, stream>>>(in, out, bn_b + (long)j * Mb * 2, Mb);
      float* t = in; in = out; out = t;
    }
    inv_kernel<6><<<blk(N), TB, 0, stream>>>(os_b, ws_b, in, INVB, a_b, N);
  }
  zero_kernel<<<blk(Ms + 1), TB, 0, stream>>>(VA, (long)(Ms + 1));
  splat_ones_kernel<3><<<blk(N), TB, 0, stream>>>(os_s, ws_s, VA, N);
  {
    float* in = VA; float* out = VB;
    for (int j = 0; j < 3; ++j) {
      blur_kernel<1><<<blk((long)Ms + 1), TB, 0, stream>>>(in, out, bn_s + (long)j * Ms * 2, Ms);
      float* t = in; in = out; out = t;
    }
    inv_kernel<3><<<blk(N), TB, 0, stream>>>(os_s, ws_s, in, INVS, a_s, N);
  }

  // ---- Q0 = softmax(-U) ----
  softmax_init_kernel<<<blk(N), TB, 0, stream>>>(unary, Q, N);

  // ---- mean-field iterations ----
  for (int it = 0; it < 5; ++it) {
    // bilateral filter of Q
    zero_kernel<<<blk((long)(Mb + 1) * C_CH), TB, 0, stream>>>(VA, (long)(Mb + 1) * C_CH);
    splat_kernel<6><<<blk((long)N * C_CH), TB, 0, stream>>>(Q, os_b, ws_b, VA, N);
    float* in = VA; float* out = VB;
    for (int j = 0; j < 6; ++j) {
      blur_kernel<C_CH><<<blk((long)(Mb + 1) * C_CH), TB, 0, stream>>>(in, out, bn_b + (long)j * Mb * 2, Mb);
      float* t = in; in = out; out = t;
    }
    slice_kernel<6, false><<<blk((long)N * C_CH), TB, 0, stream>>>(os_b, ws_b, in, INVB, 10.0f * a_b, MSG, N);

    // spatial filter of Q
    zero_kernel<<<blk((long)(Ms + 1) * C_CH), TB, 0, stream>>>(VA, (long)(Ms + 1) * C_CH);
    splat_kernel<3><<<blk((long)N * C_CH), TB, 0, stream>>>(Q, os_s, ws_s, VA, N);
    in = VA; out = VB;
    for (int j = 0; j < 3; ++j) {
      blur_kernel<C_CH><<<blk((long)(Ms + 1) * C_CH), TB, 0, stream>>>(in, out, bn_s + (long)j * Ms * 2, Ms);
      float* t = in; in = out; out = t;
    }
    slice_kernel<3, true><<<blk((long)N * C_CH), TB, 0, stream>>>(os_s, ws_s, in, INVS, 3.0f * a_s, MSG, N);

    // compatibility (WMMA) + softmax; final iteration writes d_out directly
    float* qdst = (it == 4) ? (float*)d_out : Q;
    int waves = N / 16;
    compat_softmax_wmma_kernel<<<(waves + 7) / 8, 256, 0, stream>>>(unary, MSG, qdst, N);
  }
}